// SingleGraphModel_39728447488103
// MI455X (gfx1250) — compile-verified
//
#include <hip/hip_runtime.h>
#include <hip/hip_bf16.h>
#include <stdint.h>

typedef float v2f __attribute__((ext_vector_type(2)));
typedef float v8f __attribute__((ext_vector_type(8)));

#define HIDD 64
#define HEADS 4
#define OUTD 16
#define NCLS 10

// Monotonic float -> uint key so atomicMax(u32) implements segment_max over floats.
__device__ __forceinline__ unsigned fkey(float f) {
  unsigned b = __float_as_uint(f);
  return (b & 0x80000000u) ? ~b : (b | 0x80000000u);
}
__device__ __forceinline__ float finv(unsigned k) {
  unsigned b = (k & 0x80000000u) ? (k ^ 0x80000000u) : ~k;
  return __uint_as_float(b);
}

// Swizzled LDS fill: store W[K x 64] so that for each (kstep, coltile) the
// 32 lanes' WMMA-B v2f fragments are contiguous -> single ds_load_b64 per
// fragment, bank-conflict-free, no VGPR repacking.
// Entry index: (kstep*4 + t)*32 + lane, value = {W[(4k+kh)*64+c], W[(4k+kh+1)*64+c]}
// with kh = 2*(lane>>4), c = t*16 + (lane&15).
template <int K>
__device__ __forceinline__ void fill_bswz(v2f* sW, const float* __restrict__ W) {
  for (int idx = threadIdx.x; idx < K * 32; idx += 256) {
    const int lane = idx & 31;
    const int t = (idx >> 5) & 3;
    const int kstep = idx >> 7;
    const int m = lane & 15;
    const int kh = (lane >> 4) << 1;
    const int krow = kstep * 4 + kh;
    const int c = t * 16 + m;
    v2f w;
    w.x = W[krow * HIDD + c];
    w.y = W[(krow + 1) * HIDD + c];
    sW[idx] = w;
  }
}

// ---------------------------------------------------------------------------
// Generic fp32 WMMA GEMM: Y[nrows x 64] = act(A[nrows x K] @ W[K x 64] + bias)
// One wave computes a 16x64 tile via 4 accumulators of v_wmma_f32_16x16x4_f32.
// 8 waves per block -> 128 rows/block. W staged (pre-swizzled) in LDS.
// Partial tiles: clamp row (A row m only feeds D row m; clamped rows unstored).
// ---------------------------------------------------------------------------
template <int K, int ACT, int HASB>
__global__ __launch_bounds__(256) void gemm_wmma_k(
    const float* __restrict__ A, const float* __restrict__ W,
    const float* __restrict__ bias, float* __restrict__ Y, int nrows) {
  __shared__ v2f sW[K * 32];
  fill_bswz<K>(sW, W);
  __syncthreads();

  const int lane = threadIdx.x & 31;
  const int wave = threadIdx.x >> 5;
  const int row0 = (blockIdx.x * 8 + wave) * 16;
  if (row0 >= nrows) return;  // wave-uniform: EXEC stays all-1s for WMMA

  const int m = lane & 15;
  const int kh = (lane >> 4) << 1;  // lanes 16-31 hold K+2,K+3
  const int row = min(row0 + m, nrows - 1);
  const float* arow = A + (size_t)row * K;

  v8f acc[4] = {};
  for (int k = 0; k < K; k += 4) {
    const v2f a = *(const v2f*)(arow + k + kh);
    if (k + 32 < K) __builtin_prefetch(arow + k + 32, 0, 3);
    const v2f* bs = &sW[(k >> 2) * 128];
#pragma unroll
    for (int t = 0; t < 4; ++t) {
      const v2f b = bs[t * 32 + lane];
      acc[t] = __builtin_amdgcn_wmma_f32_16x16x4_f32(
          false, a, false, b, (short)0, acc[t], false, false);
    }
  }

  const int rbase = row0 + ((lane >> 4) << 3);  // D: vgpr j = row j / j+8
#pragma unroll
  for (int t = 0; t < 4; ++t) {
    const int c = t * 16 + m;
    const float bv = HASB ? bias[c] : 0.0f;
#pragma unroll
    for (int j = 0; j < 8; ++j) {
      const int r = rbase + j;
      if (r < nrows) {
        float v = acc[t][j] + bv;
        if (ACT == 1) v = v > 0.0f ? v : 0.0f;  // relu
        Y[(size_t)r * HIDD + c] = v;
      }
    }
  }
}

// ---------------------------------------------------------------------------
// Dual GEMM: x_l = h0@W_l+b_l ; x_r = h0@W_r+b_r. Reuses the A fragment for
// both weight matrices (8 WMMA accumulators per wave).
// ---------------------------------------------------------------------------
__global__ __launch_bounds__(256) void gemm_lr_k(
    const float* __restrict__ h0, const float* __restrict__ Wl,
    const float* __restrict__ bl, const float* __restrict__ Wr,
    const float* __restrict__ br, float* __restrict__ xl,
    float* __restrict__ xr, int nrows) {
  __shared__ v2f sWl[HIDD * 32];
  __shared__ v2f sWr[HIDD * 32];
  fill_bswz<HIDD>(sWl, Wl);
  fill_bswz<HIDD>(sWr, Wr);
  __syncthreads();

  const int lane = threadIdx.x & 31;
  const int wave = threadIdx.x >> 5;
  const int row0 = (blockIdx.x * 8 + wave) * 16;
  if (row0 >= nrows) return;

  const int m = lane & 15;
  const int kh = (lane >> 4) << 1;
  const int row = min(row0 + m, nrows - 1);
  const float* arow = h0 + (size_t)row * HIDD;

  v8f accl[4] = {};
  v8f accr[4] = {};
  for (int k = 0; k < HIDD; k += 4) {
    const v2f a = *(const v2f*)(arow + k + kh);
    const v2f* bsl = &sWl[(k >> 2) * 128];
    const v2f* bsr = &sWr[(k >> 2) * 128];
#pragma unroll
    for (int t = 0; t < 4; ++t) {
      const v2f bl_ = bsl[t * 32 + lane];
      accl[t] = __builtin_amdgcn_wmma_f32_16x16x4_f32(
          false, a, false, bl_, (short)0, accl[t], false, false);
      const v2f br_ = bsr[t * 32 + lane];
      accr[t] = __builtin_amdgcn_wmma_f32_16x16x4_f32(
          false, a, false, br_, (short)0, accr[t], false, false);
    }
  }

  const int rbase = row0 + ((lane >> 4) << 3);
#pragma unroll
  for (int t = 0; t < 4; ++t) {
    const int c = t * 16 + m;
    const float bvl = bl[c];
    const float bvr = br[c];
#pragma unroll
    for (int j = 0; j < 8; ++j) {
      const int r = rbase + j;
      if (r < nrows) {
        xl[(size_t)r * HIDD + c] = accl[t][j] + bvl;
        xr[(size_t)r * HIDD + c] = accr[t][j] + bvr;
      }
    }
  }
}

// ---------------------------------------------------------------------------
// Edge-degree stats for the self-loop attribute: segment_sum(ew) & count.
// ---------------------------------------------------------------------------
__global__ void edge_deg_k(const long long* __restrict__ ei,
                           const float* __restrict__ ew,
                           float* __restrict__ sums, float* __restrict__ cnt,
                           int E) {
  int e = blockIdx.x * 256 + threadIdx.x;
  if (e >= E) return;
  int d = (int)ei[(size_t)E + e];
  atomicAdd(&sums[d], ew[e]);
  atomicAdd(&cnt[d], 1.0f);
}

__global__ void loop_attr_k(const float* __restrict__ sums,
                            const float* __restrict__ cnt,
                            float* __restrict__ lat, int n) {
  int i = blockIdx.x * 256 + threadIdx.x;
  if (i < n) lat[i] = sums[i] / fmaxf(cnt[i], 1.0f);
}

// ---------------------------------------------------------------------------
// Per-edge attention logits + running segment max (atomic u32 on ordered key).
// Edges [0,E) are real edges; [E, E+N) are self-loops with loop_attr.
// ---------------------------------------------------------------------------
__global__ void edge_alpha_k(const long long* __restrict__ ei,
                             const float* __restrict__ ew,
                             const float* __restrict__ lat,
                             const float* __restrict__ xl,
                             const float* __restrict__ xr,
                             const float* __restrict__ W_edge,
                             const float* __restrict__ att,
                             float* __restrict__ alpha,
                             unsigned* __restrict__ amax, int E, int N) {
  int e = blockIdx.x * 256 + threadIdx.x;
  if (e >= E + N) return;
  int s, d;
  float ea;
  if (e < E) {
    s = (int)ei[e];
    d = (int)ei[(size_t)E + e];
    ea = ew[e];
  } else {
    s = d = e - E;
    ea = lat[s];
  }
  const float* xls = xl + (size_t)s * HIDD;
  const float* xrd = xr + (size_t)d * HIDD;
#pragma unroll
  for (int h = 0; h < HEADS; ++h) {
    float acc = 0.0f;
#pragma unroll
    for (int o = 0; o < OUTD; ++o) {
      const int j = h * OUTD + o;
      float v = xls[j] + xrd[j] + ea * W_edge[j];
      v = v > 0.0f ? v : 0.2f * v;  // leaky_relu(0.2)
      acc += v * att[j];
    }
    alpha[(size_t)e * HEADS + h] = acc;
    atomicMax(&amax[(size_t)d * HEADS + h], fkey(acc));
  }
}

// exp(alpha - amax[dst]) -> overwrite alpha buffer; accumulate denominators.
__global__ void edge_expsum_k(const long long* __restrict__ ei,
                              float* __restrict__ alpha,
                              const unsigned* __restrict__ amax,
                              float* __restrict__ denom, int E, int N) {
  int e = blockIdx.x * 256 + threadIdx.x;
  if (e >= E + N) return;
  int d = (e < E) ? (int)ei[(size_t)E + e] : (e - E);
#pragma unroll
  for (int h = 0; h < HEADS; ++h) {
    float am = finv(amax[(size_t)d * HEADS + h]);
    float ex = __expf(alpha[(size_t)e * HEADS + h] - am);
    alpha[(size_t)e * HEADS + h] = ex;
    atomicAdd(&denom[(size_t)d * HEADS + h], ex);
  }
}

// Un-normalized numerator scatter: num[dst] += ex * x_l[src]. One thread per
// (edge, feature) -> fully coalesced gathers, fp32 atomics resolved in L2.
__global__ void edge_accum_k(const long long* __restrict__ ei,
                             const float* __restrict__ exb,
                             const float* __restrict__ xl,
                             float* __restrict__ num, int E, int N) {
  long long t = (long long)blockIdx.x * 256 + threadIdx.x;
  if (t >= (long long)(E + N) * HIDD) return;
  int e = (int)(t >> 6);
  int j = (int)(t & 63);
  int s, d;
  if (e < E) {
    s = (int)ei[e];
    d = (int)ei[(size_t)E + e];
  } else {
    s = d = e - E;
  }
  float ex = exb[(size_t)e * HEADS + (j >> 4)];
  atomicAdd(&num[(size_t)d * HIDD + j], ex * xl[(size_t)s * HIDD + j]);
}

// h = h0 + elu(num/denom + gat_bias); writes h into d_out[0 : N*64].
__global__ void node_final_k(const float* __restrict__ num,
                             const float* __restrict__ denom,
                             const float* __restrict__ gatb,
                             const float* __restrict__ h0,
                             float* __restrict__ hout, int n) {
  long long t = (long long)blockIdx.x * 256 + threadIdx.x;
  if (t >= (long long)n * HIDD) return;
  int i = (int)(t >> 6);
  int j = (int)(t & 63);
  float o = num[t] / denom[(size_t)i * HEADS + (j >> 4)] + gatb[j];
  float h1 = o > 0.0f ? o : (__expf(o) - 1.0f);  // elu
  hout[t] = h0[t] + h1;
}

// logits = z @ W2 + b2 (64 x 10): tiny K and N, plain VALU with W2 in LDS.
__global__ __launch_bounds__(256) void logits_k(const float* __restrict__ z,
                                                const float* __restrict__ W2,
                                                const float* __restrict__ b2,
                                                float* __restrict__ out, int n) {
  __shared__ float sW[HIDD * NCLS];
  __shared__ float sb[NCLS];
  for (int i = threadIdx.x; i < HIDD * NCLS; i += 256) sW[i] = W2[i];
  if (threadIdx.x < NCLS) sb[threadIdx.x] = b2[threadIdx.x];
  __syncthreads();
  int i = blockIdx.x * 256 + threadIdx.x;
  if (i >= n) return;
  const float* zr = z + (size_t)i * HIDD;
  float acc[NCLS];
#pragma unroll
  for (int c = 0; c < NCLS; ++c) acc[c] = sb[c];
  for (int k = 0; k < HIDD; ++k) {
    float zv = zr[k];
#pragma unroll
    for (int c = 0; c < NCLS; ++c) acc[c] += zv * sW[k * NCLS + c];
  }
#pragma unroll
  for (int c = 0; c < NCLS; ++c) out[(size_t)i * NCLS + c] = acc[c];
}

// ---------------------------------------------------------------------------
extern "C" void kernel_launch(void* const* d_in, const int* in_sizes, int n_in,
                              void* d_out, int out_size, void* d_ws,
                              size_t ws_size, hipStream_t stream) {
  (void)n_in; (void)out_size; (void)ws_size;
  const float* x      = (const float*)d_in[0];
  const float* ew     = (const float*)d_in[1];
  const float* W_pre  = (const float*)d_in[2];
  const float* W_l    = (const float*)d_in[3];
  const float* b_l    = (const float*)d_in[4];
  const float* W_r    = (const float*)d_in[5];
  const float* b_r    = (const float*)d_in[6];
  const float* att    = (const float*)d_in[7];
  const float* W_edge = (const float*)d_in[8];
  const float* gatb   = (const float*)d_in[9];
  const float* W1     = (const float*)d_in[10];
  const float* b1     = (const float*)d_in[11];
  const float* W2     = (const float*)d_in[12];
  const float* b2     = (const float*)d_in[13];
  const long long* ei = (const long long*)d_in[14];  // int64 (2,E) row-major

  const int N = in_sizes[0] / 128;
  const int E = in_sizes[1];
  const int EN = E + N;

  // Workspace layout (fp32). num aliases xr (dead after edge_alpha);
  // z aliases xl (dead after edge_accum).
  float* ws    = (float*)d_ws;
  float* h0    = ws;                          // N*64
  float* xl    = h0 + (size_t)N * HIDD;       // N*64 (reused as z)
  float* xr    = xl + (size_t)N * HIDD;       // N*64 (reused as num)
  float* sums  = xr + (size_t)N * HIDD;       // N
  float* cnt   = sums + (size_t)N;            // N
  float* lat   = cnt + (size_t)N;             // N
  float* denom = lat + (size_t)N;             // N*4
  unsigned* amax = (unsigned*)(denom + (size_t)N * HEADS);  // N*4
  float* alpha = (float*)(amax + (size_t)N * HEADS);        // (E+N)*4
  float* num   = xr;
  float* z     = xl;
  float* hout  = (float*)d_out;               // N*64
  float* lout  = hout + (size_t)N * HIDD;     // N*10

  // Zero sums/cnt/lat/denom/amax (contiguous 11N dwords; key 0 == float min).
  hipMemsetAsync(sums, 0, (size_t)N * 11 * sizeof(float), stream);

  const dim3 blk(256);
  const int gblocks = (N + 127) / 128;

  gemm_wmma_k<128, 0, 0><<<gblocks, blk, 0, stream>>>(x, W_pre, nullptr, h0, N);
  edge_deg_k<<<(E + 255) / 256, blk, 0, stream>>>(ei, ew, sums, cnt, E);
  loop_attr_k<<<(N + 255) / 256, blk, 0, stream>>>(sums, cnt, lat, N);
  gemm_lr_k<<<gblocks, blk, 0, stream>>>(h0, W_l, b_l, W_r, b_r, xl, xr, N);
  edge_alpha_k<<<(EN + 255) / 256, blk, 0, stream>>>(ei, ew, lat, xl, xr,
                                                     W_edge, att, alpha, amax,
                                                     E, N);
  edge_expsum_k<<<(EN + 255) / 256, blk, 0, stream>>>(ei, alpha, amax, denom,
                                                      E, N);
  hipMemsetAsync(num, 0, (size_t)N * HIDD * sizeof(float), stream);
  const long long tot = (long long)EN * HIDD;
  edge_accum_k<<<(unsigned)((tot + 255) / 256), blk, 0, stream>>>(ei, alpha, xl,
                                                                  num, E, N);
  node_final_k<<<(unsigned)(((long long)N * HIDD + 255) / 256), blk, 0,
                stream>>>(num, denom, gatb, h0, hout, N);
  gemm_wmma_k<64, 1, 1><<<gblocks, blk, 0, stream>>>(hout, W1, b1, z, N);
  logits_k<<<(N + 255) / 256, blk, 0, stream>>>(z, W2, b2, lout, N);
}